// GaussianKernels_84731114816366
// MI455X (gfx1250) — compile-verified
//
#include <hip/hip_runtime.h>
#include <hip/hip_bf16.h>

// CDNA5 / gfx1250: wave32, WMMA 16x16x4 F32, async global->LDS gather staging,
// LDS histogram scatter (ds_add_f32).
typedef __attribute__((ext_vector_type(2))) float v2f;
typedef __attribute__((ext_vector_type(8))) float v8f;

#define Ddim   128
#define Kn     200
#define NC     1000
#define BLOCK  128
#define NWAVES 4

// Dynamic-LDS layout (float indices). No static __shared__ anywhere, so the
// dynamic arena starts at LDS offset 0 and these are also the raw byte/4
// offsets used by the async-copy instructions.
#define F_OFF    0                    // [0,128)    feature vector
#define HIST_OFF 128                  // [128,1128) class histogram
#define RED_OFF  1128                 // [1128,1256) block-reduce scratch
#define DIAG_OFF 1256                 // [1256,2280) 256 floats per wave (diag dump)
#define TILE_OFF 2304                 // [2304,10496) 2048 floats per wave (16 rows x 128)
#define SMEM_FLOATS (TILE_OFF + NWAVES * 16 * Ddim)   // 10496 floats = 41984 B

__global__ __launch_bounds__(BLOCK)
void GaussianKernels_84731114816366_kernel(
    const float* __restrict__ features,   // [B, 128]
    const float* __restrict__ centres,    // [100000, 128]
    const float* __restrict__ weight,     // [100000]
    const int*   __restrict__ labels,     // [100000]
    const int*   __restrict__ neighbours, // [B, 200]
    float*       __restrict__ out)        // [B, 1000]
{
    extern __shared__ float smem[];
    float* s_f    = smem + F_OFF;
    float* s_hist = smem + HIST_OFF;
    float* s_red  = smem + RED_OFF;

    const int b    = blockIdx.x;
    const int tid  = threadIdx.x;
    const int lane = tid & 31;
    const int wave = tid >> 5;

    // ---- init: feature vector + zeroed histogram ----
    if (tid < Ddim) s_f[tid] = features[(size_t)b * Ddim + tid];
    for (int c = tid; c < NC; c += BLOCK) s_hist[c] = 0.0f;
    __syncthreads();

    const int* nbr  = neighbours + (size_t)b * Kn;
    float*     tile = smem + TILE_OFF + wave * (16 * Ddim);
    float*     diag = smem + DIAG_OFF + wave * 256;
    const unsigned tileByte = (unsigned)(TILE_OFF + wave * (16 * Ddim)) * 4u;
    const unsigned long long cbase = (unsigned long long)centres;

    // Per-lane A/B operand dims: lanes 0-15 take {4s,4s+1}, lanes 16-31 {4s+2,4s+3}.
    const int rowA   = lane & 15;
    const int colOff = (lane >> 4) << 1;

    // Preload this lane's 64 feature operands into registers (as 32 float2) so
    // the asm "memory" clobbers below cannot force s_f reloads in the tile loop.
    v2f fr[32];
    #pragma unroll
    for (int s = 0; s < 32; ++s)
        fr[s] = *(const v2f*)(s_f + s * 4 + colOff);   // 8B-aligned LDS load

    const int NT = (Kn + 15) / 16; // 13 tiles of 16 neighbours
    for (int t = wave; t < NT; t += NWAVES) {
        const int k0 = t * 16;

        // ---- Async gather: 16 centre rows -> LDS, no VGPR round-trip.
        // Each row is one coalesced 512B transfer (32 lanes x b128), table is
        // L2-resident (51MB < 192MB L2). Tracked by ASYNCcnt.
        for (int r = 0; r < 16; ++r) {
            int k = k0 + r; if (k > Kn - 1) k = Kn - 1;   // clamp tail, masked later
            const int idx = nbr[k];
            const unsigned voff = (unsigned)idx * (Ddim * 4u) + (unsigned)lane * 16u;
            const unsigned dst  = tileByte + (unsigned)r * 512u + (unsigned)lane * 16u;
            asm volatile("global_load_async_to_lds_b128 %0, %1, %2"
                         :: "v"(dst), "v"(voff), "s"(cbase)
                         : "memory");
        }
        asm volatile("s_wait_asynccnt 0x0" ::: "memory");

        // ---- Distances for 16 neighbours via diag(E * E^T), E = f - c (16x128).
        // The f32 16x4 A-layout and 4x16 B-layout coincide for E vs E^T, so one
        // float2 per lane serves as BOTH SRC0 and SRC1 of the WMMA.
        const float* rp = tile + rowA * Ddim;
        v8f acc = {};
        #pragma unroll
        for (int s = 0; s < 32; ++s) {
            const v2f c2 = *(const v2f*)(rp + s * 4 + colOff); // 8B-aligned -> b64 LDS load
            const v2f e  = fr[s] - c2;                         // v_pk_add_f32 neg
            acc = __builtin_amdgcn_wmma_f32_16x16x4_f32(
                      false, e, false, e, (short)0, acc, false, false);
        }

        // ---- Extract diagonal: dump C/D layout (VGPR r: M=r lanes0-15,
        // M=r+8 lanes16-31) into this wave's private diag scratch.
        #pragma unroll
        for (int r = 0; r < 8; ++r) diag[r * 32 + lane] = acc[r];

        if (lane < 16) {
            const int   m    = lane;
            const float dist = (m < 8) ? diag[m * 32 + m]
                                       : diag[(m - 8) * 32 + (m + 16)];
            const int k = k0 + m;
            if (k < Kn) {
                const int idx = nbr[k];
                // exp(-d/(2*sigma^2)) * exp(w) = exp(w - 0.5*d), sigma = 1
                const float kern = __expf(weight[idx] - 0.5f * dist);
                atomicAdd(&s_hist[labels[idx]], kern);   // ds_add_f32
            }
        }
    }
    __syncthreads();

    // ---- epsilon-fill, normalize, log ----
    float lsum = 0.0f;
    for (int c = tid; c < NC; c += BLOCK) {
        float p = s_hist[c];
        p = (p == 0.0f) ? 1e-10f : p;
        s_hist[c] = p;
        lsum += p;
    }
    s_red[tid] = lsum;
    __syncthreads();
    for (int off = BLOCK / 2; off > 0; off >>= 1) {
        if (tid < off) s_red[tid] += s_red[tid + off];
        __syncthreads();
    }
    const float logTot = __logf(s_red[0]);

    float* po = out + (size_t)b * NC;
    for (int c = tid; c < NC; c += BLOCK)
        po[c] = __logf(s_hist[c]) - logTot;
}

extern "C" void kernel_launch(void* const* d_in, const int* in_sizes, int n_in,
                              void* d_out, int out_size, void* d_ws, size_t ws_size,
                              hipStream_t stream) {
    const float* features   = (const float*)d_in[0];
    const float* centres    = (const float*)d_in[1];
    const float* weight     = (const float*)d_in[2];
    const int*   labels     = (const int*)d_in[3];
    const int*   neighbours = (const int*)d_in[4];
    float*       out        = (float*)d_out;

    const int B = in_sizes[0] / Ddim;  // 4096
    GaussianKernels_84731114816366_kernel<<<B, BLOCK, SMEM_FLOATS * 4, stream>>>(
        features, centres, weight, labels, neighbours, out);
}